// TimeStepGCN_47949014892560
// MI455X (gfx1250) — compile-verified
//
#include <hip/hip_runtime.h>

typedef __attribute__((ext_vector_type(2))) float v2f;
typedef __attribute__((ext_vector_type(8))) float v8f;

#define HDIM 128

// ---------------- degree / dinv ----------------
__global__ __launch_bounds__(256) void k_deg_init(float* __restrict__ deg, int N) {
    int i = blockIdx.x * 256 + threadIdx.x;
    if (i < N) deg[i] = 1.0f;                     // self loop
}

__global__ __launch_bounds__(256) void k_deg_accum(float* __restrict__ deg,
                                                   const int* __restrict__ col, int E) {
    int e = blockIdx.x * 256 + threadIdx.x;
    if (e < E) atomicAdd(&deg[col[e]], 1.0f);
}

__global__ __launch_bounds__(256) void k_dinv(float* __restrict__ deg, int N) {
    int i = blockIdx.x * 256 + threadIdx.x;
    if (i < N) deg[i] = rsqrtf(deg[i]);           // deg >= 1 always (self loops)
}

// ---------------- WMMA GEMM:  Y[N x 128] = X[N x K] @ W[K x 128] ----------------
// One block = 8 waves = 128 rows.  Each wave: 16 rows x 128 cols (8 v8f accumulators).
template <int K>
__global__ __launch_bounds__(256) void k_gemm_wmma(const float* __restrict__ X,
                                                   const float* __restrict__ W,
                                                   float* __restrict__ Y) {
    constexpr int LDSK = K + 2;                   // pad: stride%64 == 2 -> conflict-free
    __shared__ float sA[128 * LDSK];

    const int tid = threadIdx.x;
    const size_t block_m0 = (size_t)blockIdx.x * 128;

    // cooperative coalesced load of the 128 x K X-tile
    constexpr int TOT4 = 128 * K / 4;
    for (int i = tid; i < TOT4; i += 256) {
        const int r  = i / (K / 4);
        const int c4 = i % (K / 4);
        const float4 v = reinterpret_cast<const float4*>(X + (block_m0 + r) * K)[c4];
        float* dst = sA + r * LDSK + c4 * 4;
        dst[0] = v.x; dst[1] = v.y; dst[2] = v.z; dst[3] = v.w;
    }
    __syncthreads();

    const int wave  = tid >> 5;
    const int lane  = tid & 31;
    const int m0    = wave * 16;                  // row offset inside block
    const int rowA  = m0 + (lane & 15);
    const int khalf = (lane >> 4) * 2;            // hi half-wave handles K+2,K+3
    const int colB  = lane & 15;

    v8f acc[8];
#pragma unroll
    for (int nt = 0; nt < 8; ++nt) acc[nt] = (v8f){0.f,0.f,0.f,0.f,0.f,0.f,0.f,0.f};

    for (int kt = 0; kt < K / 4; ++kt) {
        const int kb = kt * 4 + khalf;
        v2f a;
        a.x = sA[rowA * LDSK + kb];
        a.y = sA[rowA * LDSK + kb + 1];
        const float* Wk = W + (size_t)kb * HDIM;
#pragma unroll
        for (int nt = 0; nt < 8; ++nt) {
            v2f b;
            b.x = Wk[nt * 16 + colB];
            b.y = Wk[HDIM + nt * 16 + colB];
            // D = A(16x4,f32) * B(4x16,f32) + C   ->  v_wmma_f32_16x16x4_f32
            acc[nt] = __builtin_amdgcn_wmma_f32_16x16x4_f32(
                false, a, false, b, (short)0, acc[nt], false, false);
        }
    }

    // C/D layout: VGPR v holds M=v (lanes 0-15) and M=v+8 (lanes 16-31); N = lane&15
    const size_t gm = block_m0 + m0 + (lane >> 4) * 8;
#pragma unroll
    for (int nt = 0; nt < 8; ++nt)
#pragma unroll
        for (int v = 0; v < 8; ++v)
            Y[(gm + v) * HDIM + nt * 16 + colB] = acc[nt][v];
}

// ---------------- aggregation ----------------
// OUT[i,:] = XW[i,:] * dinv[i]^2   (self-loop term; also initializes OUT)
__global__ __launch_bounds__(256) void k_agg_self(const float* __restrict__ XW,
                                                  float* __restrict__ OUT,
                                                  const float* __restrict__ dinv, int N) {
    int i = blockIdx.x * 256 + threadIdx.x;       // one float4 per thread
    if (i >= N * (HDIM / 4)) return;
    int node = i >> 5;
    float d = dinv[node];
    float w = d * d;
    float4 v = reinterpret_cast<const float4*>(XW)[i];
    v.x *= w; v.y *= w; v.z *= w; v.w *= w;
    reinterpret_cast<float4*>(OUT)[i] = v;
}

// wave-per-edge: OUT[c,:] += XW[r,:] * dinv[r]*dinv[c]   (fp32 global atomics, L2-resident)
__global__ __launch_bounds__(256) void k_agg_edges(const float* __restrict__ XW,
                                                   float* __restrict__ OUT,
                                                   const int* __restrict__ row,
                                                   const int* __restrict__ col,
                                                   const float* __restrict__ dinv, int E) {
    int g = blockIdx.x * 256 + threadIdx.x;
    int e = g >> 5;
    int lane = g & 31;
    if (e >= E) return;
    int r = row[e], c = col[e];
    float w = dinv[r] * dinv[c];
    const float4 v = reinterpret_cast<const float4*>(XW + (size_t)r * HDIM)[lane];
    float* dst = OUT + (size_t)c * HDIM + lane * 4;
    atomicAdd(dst + 0, v.x * w);
    atomicAdd(dst + 1, v.y * w);
    atomicAdd(dst + 2, v.z * w);
    atomicAdd(dst + 3, v.w * w);
}

// H = relu(H + bias), in place (float4 per thread)
__global__ __launch_bounds__(256) void k_bias_relu(float* __restrict__ H,
                                                   const float* __restrict__ bias, int N) {
    int i = blockIdx.x * 256 + threadIdx.x;
    if (i >= N * (HDIM / 4)) return;
    int chunk = i & 31;
    float4 v = reinterpret_cast<float4*>(H)[i];
    float4 b = reinterpret_cast<const float4*>(bias)[chunk];
    v.x = fmaxf(v.x + b.x, 0.f); v.y = fmaxf(v.y + b.y, 0.f);
    v.z = fmaxf(v.z + b.z, 0.f); v.w = fmaxf(v.w + b.w, 0.f);
    reinterpret_cast<float4*>(H)[i] = v;
}

// wave-per-node: h = LN(relu(A + bias)) * g + b   (wave32 shuffle reductions)
__global__ __launch_bounds__(256) void k_bias_relu_ln(const float* __restrict__ A,
                                                      const float* __restrict__ bias,
                                                      const float* __restrict__ g,
                                                      const float* __restrict__ bln,
                                                      float* __restrict__ Hout, int N) {
    int gid = blockIdx.x * 256 + threadIdx.x;
    int n = gid >> 5;
    int lane = gid & 31;
    if (n >= N) return;
    float4 v = reinterpret_cast<const float4*>(A + (size_t)n * HDIM)[lane];
    float4 bb = reinterpret_cast<const float4*>(bias)[lane];
    v.x = fmaxf(v.x + bb.x, 0.f); v.y = fmaxf(v.y + bb.y, 0.f);
    v.z = fmaxf(v.z + bb.z, 0.f); v.w = fmaxf(v.w + bb.w, 0.f);

    float s = v.x + v.y + v.z + v.w;
#pragma unroll
    for (int m = 16; m >= 1; m >>= 1) s += __shfl_xor(s, m, 32);
    float mu = s * (1.0f / HDIM);

    float4 d = {v.x - mu, v.y - mu, v.z - mu, v.w - mu};
    float q = d.x * d.x + d.y * d.y + d.z * d.z + d.w * d.w;
#pragma unroll
    for (int m = 16; m >= 1; m >>= 1) q += __shfl_xor(q, m, 32);
    float rs = rsqrtf(q * (1.0f / HDIM) + 1e-5f);

    float4 gg = reinterpret_cast<const float4*>(g)[lane];
    float4 lb = reinterpret_cast<const float4*>(bln)[lane];
    float4 o = {d.x * rs * gg.x + lb.x, d.y * rs * gg.y + lb.y,
                d.z * rs * gg.z + lb.z, d.w * rs * gg.w + lb.w};
    reinterpret_cast<float4*>(Hout + (size_t)n * HDIM)[lane] = o;
}

// thread-per-node Q heads (Wq1/Wq2 staged in LDS): q = h @ Wq + bq, A=5 each
__global__ __launch_bounds__(256) void k_qheads(const float* __restrict__ Hf,
                                                const float* __restrict__ Wq1,
                                                const float* __restrict__ bq1,
                                                const float* __restrict__ Wq2,
                                                const float* __restrict__ bq2,
                                                float* __restrict__ q1,
                                                float* __restrict__ q2, int N) {
    __shared__ float sW[HDIM * 5 * 2];
    for (int i = threadIdx.x; i < HDIM * 5; i += 256) {
        sW[i] = Wq1[i];
        sW[HDIM * 5 + i] = Wq2[i];
    }
    __syncthreads();
    int n = blockIdx.x * 256 + threadIdx.x;
    if (n >= N) return;
    float a1[5], a2[5];
#pragma unroll
    for (int a = 0; a < 5; ++a) { a1[a] = bq1[a]; a2[a] = bq2[a]; }
    const float* hrow = Hf + (size_t)n * HDIM;
    for (int f = 0; f < HDIM; ++f) {
        float hv = hrow[f];
#pragma unroll
        for (int a = 0; a < 5; ++a) {
            a1[a] = fmaf(hv, sW[f * 5 + a], a1[a]);
            a2[a] = fmaf(hv, sW[HDIM * 5 + f * 5 + a], a2[a]);
        }
    }
#pragma unroll
    for (int a = 0; a < 5; ++a) {
        q1[(size_t)n * 5 + a] = a1[a];
        q2[(size_t)n * 5 + a] = a2[a];
    }
}

extern "C" void kernel_launch(void* const* d_in, const int* in_sizes, int n_in,
                              void* d_out, int out_size, void* d_ws, size_t ws_size,
                              hipStream_t stream) {
    const float* x    = (const float*)d_in[0];
    const int*   ei   = (const int*)d_in[1];
    const float* W1   = (const float*)d_in[2];
    const float* b1   = (const float*)d_in[3];
    const float* W2   = (const float*)d_in[4];
    const float* b2   = (const float*)d_in[5];
    const float* ln_g = (const float*)d_in[6];
    const float* ln_b = (const float*)d_in[7];
    const float* Wq1  = (const float*)d_in[8];
    const float* bq1  = (const float*)d_in[9];
    const float* Wq2  = (const float*)d_in[10];
    const float* bq2  = (const float*)d_in[11];

    const int N = in_sizes[0] / 64;   // 32768 nodes (D = 64)
    const int E = in_sizes[1] / 2;    // 524288 edges
    const int* row = ei;
    const int* col = ei + E;

    float* dinv = (float*)d_ws;
    float* bufA = dinv + N;                       // N x 128 (xw)
    float* bufB = bufA + (size_t)N * HDIM;        // N x 128 (agg / h)

    float* q1   = (float*)d_out;
    float* q2   = q1 + (size_t)N * 5;
    float* hout = q1 + (size_t)N * 10;

    const int vecBlocks  = (N * (HDIM / 4) + 255) / 256;   // float4-grain elementwise
    const int nodeWaves  = (N * 32 + 255) / 256;           // wave-per-node
    const int edgeWaves  = (E * 32 + 255) / 256;           // wave-per-edge

    // degrees -> dinv
    k_deg_init <<<(N + 255) / 256, 256, 0, stream>>>(dinv, N);
    k_deg_accum<<<(E + 255) / 256, 256, 0, stream>>>(dinv, col, E);
    k_dinv     <<<(N + 255) / 256, 256, 0, stream>>>(dinv, N);

    // layer 1
    k_gemm_wmma<64><<<N / 128, 256, 0, stream>>>(x, W1, bufA);
    k_agg_self <<<vecBlocks, 256, 0, stream>>>(bufA, bufB, dinv, N);
    k_agg_edges<<<edgeWaves, 256, 0, stream>>>(bufA, bufB, row, col, dinv, E);
    k_bias_relu<<<vecBlocks, 256, 0, stream>>>(bufB, b1, N);

    // layer 2 (bufB = h1 is consumed by the GEMM, then recycled as the agg target)
    k_gemm_wmma<128><<<N / 128, 256, 0, stream>>>(bufB, W2, bufA);
    k_agg_self <<<vecBlocks, 256, 0, stream>>>(bufA, bufB, dinv, N);
    k_agg_edges<<<edgeWaves, 256, 0, stream>>>(bufA, bufB, row, col, dinv, E);
    k_bias_relu_ln<<<nodeWaves, 256, 0, stream>>>(bufB, b2, ln_g, ln_b, hout, N);

    // heads
    k_qheads<<<(N + 255) / 256, 256, 0, stream>>>(hout, Wq1, bq1, Wq2, bq2, q1, q2, N);
}